// FastAttentionPool_59803124630036
// MI455X (gfx1250) — compile-verified
//
#include <hip/hip_runtime.h>
#include <math.h>

// FastAttentionPool on MI455X (gfx1250):
//  - one workgroup per graph segment (batch is sorted -> contiguous rows)
//  - single HBM pass over x: rows staged into LDS via CDNA5 async global->LDS
//    (non-temporal: x has zero reuse, keep it out of L2)
//  - online (streaming) softmax with running max/sum + rescaled accumulator
//  - 256 threads = 8 wave32; each thread owns output columns 2*tid, 2*tid+1

#define D_NODE   512
#define D_TGT    128
#define CHUNK    24          // rows staged per LDS tile (24*512*4 = 48 KB)
#define NTHREADS 256
#define CPOL_NT  1           // gfx12-style cpol: TH[2:0]=1 (non-temporal)

#if __has_builtin(__builtin_amdgcn_global_load_async_to_lds_b128)
#define USE_ASYNC 1
#else
#define USE_ASYNC 0
#endif

typedef int v4i __attribute__((ext_vector_type(4)));
typedef __attribute__((address_space(1))) v4i* gv4i_p;   // global (AS1)
typedef __attribute__((address_space(3))) v4i* lv4i_p;   // LDS    (AS3)

__device__ __forceinline__ void stage16(const float* g, float* l) {
#if USE_ASYNC
  __builtin_amdgcn_global_load_async_to_lds_b128(
      (gv4i_p)(void*)g, (lv4i_p)(void*)l, 0, CPOL_NT);
#else
  float4 v = *(const float4*)g;
  *(float4*)l = v;
#endif
}

__device__ __forceinline__ void wait_async_zero() {
#if USE_ASYNC
#if __has_builtin(__builtin_amdgcn_s_wait_asynccnt)
  __builtin_amdgcn_s_wait_asynccnt(0);
#else
  asm volatile("s_wait_asynccnt 0" ::: "memory");
#endif
#endif
}

__global__ __launch_bounds__(NTHREADS) void FastAttentionPool_kernel(
    const float* __restrict__ x,      // [N, 512]
    const int*   __restrict__ batch,  // [N] sorted graph ids
    const float* __restrict__ te,     // [B, 128]
    const float* __restrict__ w_x,    // [512]
    const float* __restrict__ bias,   // [1]
    const float* __restrict__ Wt,     // [128]
    float*       __restrict__ out,    // [B, 512]
    int N)
{
  __shared__ float xs[CHUNK][D_NODE];   // staged segment rows
  __shared__ float wx[D_NODE];
  __shared__ float lg[32];              // per-row logits (chunk)
  __shared__ float ev[32];              // per-row exp(logit - M)
  __shared__ float sc[4];               // 0:t_b  1:M_new  2:chunk_sum
  __shared__ int   bounds[2];

  const int tid  = threadIdx.x;
  const int lane = tid & 31;
  const int wave = tid >> 5;
  const int b    = blockIdx.x;
  const int col  = tid << 1;            // this thread's column pair

  // w_x -> LDS (broadcast operand for all row dots)
  for (int i = tid; i < D_NODE; i += NTHREADS) wx[i] = w_x[i];

  // segment bounds: lower_bound(batch,b) .. lower_bound(batch,b+1)
  if (tid == 0) {
    int lo = 0, hi = N;
    while (lo < hi) { int mid = (lo + hi) >> 1; if (batch[mid] <  b)     lo = mid + 1; else hi = mid; }
    bounds[0] = lo;
    int lo2 = lo; hi = N;
    while (lo2 < hi){ int mid = (lo2 + hi) >> 1; if (batch[mid] < b + 1) lo2 = mid + 1; else hi = mid; }
    bounds[1] = lo2;
  }

  // t_b = target_embed[b] . W_t  (wave 0: 4 elements/lane, shfl reduce)
  if (wave == 0) {
    float p = 0.f;
    #pragma unroll
    for (int k = 0; k < 4; ++k) {
      int i = lane * 4 + k;
      p += te[(size_t)b * D_TGT + i] * Wt[i];
    }
    #pragma unroll
    for (int off = 16; off; off >>= 1) p += __shfl_xor(p, off, 32);
    if (lane == 0) sc[0] = p;
  }
  __syncthreads();

  const int   seg_s = bounds[0], seg_e = bounds[1];
  const float tb    = sc[0] + bias[0];

  float m = -INFINITY, s = 0.f, acc0 = 0.f, acc1 = 0.f;

  for (int cs = seg_s; cs < seg_e; cs += CHUNK) {
    const int L = min(CHUNK, seg_e - cs);

    // ---- stage L rows of x into LDS (async global->LDS, NT, 16B packets) ----
    const int packets = L << 7;                     // L * 128 packets of 16B
    for (int p = tid; p < packets; p += NTHREADS) {
      const int r = p >> 7;
      const int c = (p & 127) << 2;                 // float column
      stage16(x + (size_t)(cs + r) * D_NODE + c, &xs[r][c]);
    }
    wait_async_zero();
    __syncthreads();

    // ---- logits: one row per wave, conflict-free strided LDS dot ----
    for (int r = wave; r < L; r += 8) {
      float p = 0.f;
      #pragma unroll
      for (int k = 0; k < 16; ++k)
        p += xs[r][lane + 32 * k] * wx[lane + 32 * k];
      #pragma unroll
      for (int off = 16; off; off >>= 1) p += __shfl_xor(p, off, 32);
      if (lane == 0) lg[r] = p + tb;
    }
    __syncthreads();

    // ---- wave 0: chunk max, exp values, chunk sum ----
    if (wave == 0) {
      float v  = (lane < L) ? lg[lane] : -INFINITY;
      float mx = v;
      #pragma unroll
      for (int off = 16; off; off >>= 1) mx = fmaxf(mx, __shfl_xor(mx, off, 32));
      const float Mn = fmaxf(m, mx);
      const float e  = (lane < L) ? __expf(v - Mn) : 0.f;
      ev[lane] = e;
      float su = e;
      #pragma unroll
      for (int off = 16; off; off >>= 1) su += __shfl_xor(su, off, 32);
      if (lane == 0) { sc[1] = Mn; sc[2] = su; }
    }
    __syncthreads();

    const float Mn    = sc[1];
    const float csum  = sc[2];
    const float scale = __expf(m - Mn);   // m == -inf on first chunk -> 0
    acc0 *= scale;
    acc1 *= scale;
    s = s * scale + csum;
    m = Mn;

    // ---- weighted accumulate from LDS (adjacent column pair -> ds_load_b64) ----
    for (int r = 0; r < L; ++r) {
      const float  e = ev[r];
      const float2 v = *(const float2*)&xs[r][col];
      acc0 = fmaf(e, v.x, acc0);
      acc1 = fmaf(e, v.y, acc1);
    }
    __syncthreads();   // xs reused next chunk
  }

  const float inv = 1.0f / (s + 1e-16f);
  float2 o; o.x = acc0 * inv; o.y = acc1 * inv;
  *(float2*)&out[(size_t)b * D_NODE + col] = o;     // single global_store_b64
}

extern "C" void kernel_launch(void* const* d_in, const int* in_sizes, int n_in,
                              void* d_out, int out_size, void* d_ws, size_t ws_size,
                              hipStream_t stream) {
  const float* x     = (const float*)d_in[0];
  const int*   batch = (const int*)  d_in[1];
  const float* te    = (const float*)d_in[2];
  const float* w_x   = (const float*)d_in[3];
  const float* bias  = (const float*)d_in[4];
  const float* Wt    = (const float*)d_in[5];
  const int N = in_sizes[0] / D_NODE;
  const int B = in_sizes[2] / D_TGT;
  float* out = (float*)d_out;

  FastAttentionPool_kernel<<<B, NTHREADS, 0, stream>>>(x, batch, te, w_x, bias, Wt, out, N);
}